// SimpleNet_89189290869163
// MI455X (gfx1250) — compile-verified
//
#include <hip/hip_runtime.h>
#include <hip/hip_bf16.h>

typedef __attribute__((ext_vector_type(2))) float v2f;
typedef __attribute__((ext_vector_type(8))) float v8f;

#define IMG  512
#define TW   32
#define TH   32
#define HALO 5
#define XSW  (TW + 2 * HALO)   // 42 staged columns
#define XSH  (TH + 2 * HALO)   // 42 staged rows
#define XSTR 44                // padded LDS stride (even, bank-friendly)
#define NT   128               // 4 waves of 32

// Normalized 1-D gaussian taps for sigma=3, m=3 (7x7 kernel = outer(g,g)).
// g[k] = exp(-(k-3)^2/18) / sum
__device__ __forceinline__ float gtap(int d) {
  int e = d - 3; e = e < 0 ? -e : e;           // |d-3|; e>3 => outside band => 0
  float r = 0.0f;
  r = (e == 3) ? 0.1062886f : r;
  r = (e == 2) ? 0.1403213f : r;
  r = (e == 1) ? 0.1657702f : r;
  r = (e == 0) ? 0.1752401f : r;
  return r;
}

__global__ __launch_bounds__(NT) void fused_stencil(
    const float* __restrict__ X,
    const float* __restrict__ w0, const float* __restrict__ b0,
    const float* __restrict__ w1, const float* __restrict__ w2,
    const float* __restrict__ w3, const float* __restrict__ w4,
    const float* __restrict__ w5, const float* __restrict__ w6,
    float* __restrict__ out)
{
  __shared__ float xs[XSH][XSTR];                 // staged input + halo (0-padded)
  __shared__ float vmn3[TH][XSTR],  vmx3[TH][XSTR];
  __shared__ float vmn11[TH][XSTR], vmx11[TH][XSTR];
  __shared__ float vg[TH][XSTR];                  // vertical 7-tap gaussian
  __shared__ float hm3mn[TH][XSTR], hm3mx[TH][XSTR];
  __shared__ float gsc[NT / 32][16][17];          // per-wave WMMA D bounce

  const int tid = threadIdx.x;
  const int tx0 = blockIdx.x * TW;
  const int ty0 = blockIdx.y * TH;
  const int bz  = blockIdx.z;
  const float INF = __builtin_huge_valf();

  const float* __restrict__ Xb = X + (size_t)bz * (IMG * IMG);

  // ---- stage input tile + halo; OOB -> 0 (correct pad for both convs) ----
  for (int p = tid; p < XSH * XSW; p += NT) {
    int y = p / XSW, x = p - y * XSW;
    int gy = ty0 + y - HALO, gx = tx0 + x - HALO;
    float v = 0.0f;
    if ((unsigned)gy < (unsigned)IMG && (unsigned)gx < (unsigned)IMG)
      v = Xb[gy * IMG + gx];
    xs[y][x] = v;
  }
  __syncthreads();

  // ---- pass 1: all vertical reductions, fused over one xs sweep ----
  for (int p = tid; p < TH * XSW; p += NT) {
    int r = p / XSW, c = p - (p / XSW) * XSW;
    int gy = ty0 + r;                 // output row, always in-range
    int gx = tx0 + c - HALO;          // staged column, may be OOB
    float mn3 = INF, mx3 = -INF, mn11 = INF, mx11 = -INF, acc = 0.0f;
#pragma unroll
    for (int dy = -5; dy <= 5; ++dy) {
      float v  = xs[r + HALO + dy][c];
      bool ok  = (unsigned)(gy + dy) < (unsigned)IMG;   // row validity for pools
      float vn = ok ? v : INF;
      float vx = ok ? v : -INF;
      mn11 = fminf(mn11, vn);  mx11 = fmaxf(mx11, vx);
      if (dy >= -1 && dy <= 1) { mn3 = fminf(mn3, vn); mx3 = fmaxf(mx3, vx); }
      if (dy >= -3 && dy <= 3) acc = fmaf(gtap(dy + 3), v, acc);  // 0-pad ok
    }
    bool cok = (unsigned)gx < (unsigned)IMG;            // column validity for pools
    vmn3[r][c]  = cok ? mn3  :  INF;
    vmx3[r][c]  = cok ? mx3  : -INF;
    vmn11[r][c] = cok ? mn11 :  INF;
    vmx11[r][c] = cok ? mx11 : -INF;
    vg[r][c]    = acc;                                  // OOB col is already 0
  }
  __syncthreads();

  // ---- pass 1.5: horizontal 3-wide pre-reduction of the 11-tap results ----
  for (int p = tid; p < TH * 40; p += NT) {
    int r = p / 40, c = 1 + (p - (p / 40) * 40);        // c in [1,41)
    hm3mn[r][c] = fminf(fminf(vmn11[r][c - 1], vmn11[r][c]), vmn11[r][c + 1]);
    hm3mx[r][c] = fmaxf(fmaxf(vmx11[r][c - 1], vmx11[r][c]), vmx11[r][c + 1]);
  }
  __syncthreads();

  // ---- pass 2: per-wave 16x16 sub-tile ----
  const int wv   = tid >> 5;        // wave 0..3
  const int ln   = tid & 31;
  const int hi   = ln >> 4;         // lane half (K split)
  const int n    = ln & 15;         // lane-within-half: row M for A, col N for B/D
  const int wsub = (wv & 1) * 16;
  const int rsub = (wv >> 1) * 16;

  // Constant banded B(24x16): B[j][w] = g[j-w] if 0<=j-w<=6 else 0.
  // f32 B layout: lanes<16 hold K=k0,k0+1 in v0,v1; lanes>=16 hold K=k0+2,k0+3.
  v2f bf[6];
#pragma unroll
  for (int s = 0; s < 6; ++s) {
    int j0 = 4 * s + 2 * hi;
    bf[s].x = gtap(j0 - n);
    bf[s].y = gtap(j0 + 1 - n);
  }

  // Horizontal 7-tap gaussian: Out(16x16) = A(16x24) x B(24x16),
  // A[M][j] = vg[rsub+M][wsub + j - 3 + HALO]; pad cols j=22,23 hit zero bands.
  v8f acc = {};
#pragma unroll
  for (int s = 0; s < 6; ++s) {
    int col = wsub + 4 * s + 2 * hi + 2;                // always even -> 8B aligned
    v2f a = *(const v2f*)&vg[rsub + n][col];
    acc = __builtin_amdgcn_wmma_f32_16x16x4_f32(
        false, a, false, bf[s], (short)0, acc, false, false);
  }
  // D layout: VGPR v -> row v (lanes<16) / v+8 (lanes>=16), col = n.
#pragma unroll
  for (int v = 0; v < 8; ++v)
    gsc[wv][v + 8 * hi][n] = acc[v];
  __syncthreads();

  // ---- per-channel scalar weights ----
  const float W0[2] = {w0[0], w0[1]}, Bc[2] = {b0[0], b0[1]};
  const float W1[2] = {w1[0], w1[1]}, W2[2] = {w2[0], w2[1]};
  const float W3[2] = {w3[0], w3[1]}, W4[2] = {w4[0], w4[1]};
  const float W5[2] = {w5[0], w5[1]}, W6[2] = {w6[0], w6[1]};

  const size_t outHW = (size_t)IMG * IMG;
  float* __restrict__ out0 = out + ((size_t)bz * 2) * outHW;
  float* __restrict__ out1 = out0 + outHW;

  // ---- combine + store; lane = column (coalesced), 8 rows per lane ----
#pragma unroll
  for (int i = 0; i < 8; ++i) {
    int lr = 2 * i + hi;              // row within sub-tile
    int ar = rsub + lr;               // row within tile arrays
    int ac = wsub + n + HALO;         // staged column of this pixel
    float xv = xs[ar + HALO][ac];
    float x1 = 0.25f * (xs[ar + HALO - 1][ac] + xs[ar + HALO + 1][ac] +
                        xs[ar + HALO][ac - 1] + xs[ar + HALO][ac + 1]);
    float x2 = gsc[wv][lr][n];
    float x3 = fminf(fminf(vmn3[ar][ac - 1], vmn3[ar][ac]), vmn3[ar][ac + 1]);
    float x5 = fmaxf(fmaxf(vmx3[ar][ac - 1], vmx3[ar][ac]), vmx3[ar][ac + 1]);
    float x4 = fminf(fminf(fminf(hm3mn[ar][ac - 4], hm3mn[ar][ac - 1]),
                           hm3mn[ar][ac + 2]),
                     fminf(vmn11[ar][ac + 4], vmn11[ar][ac + 5]));
    float x6 = fmaxf(fmaxf(fmaxf(hm3mx[ar][ac - 4], hm3mx[ar][ac - 1]),
                           hm3mx[ar][ac + 2]),
                     fmaxf(vmx11[ar][ac + 4], vmx11[ar][ac + 5]));

    int gr = ty0 + ar, gw = tx0 + wsub + n;
    size_t o = (size_t)gr * IMG + gw;

    float r0 = fmaf(W0[0], xv, Bc[0]);
    r0 = fmaf(W1[0], x1, r0); r0 = fmaf(W2[0], x2, r0);
    r0 = fmaf(W3[0], x3, r0); r0 = fmaf(W4[0], x4, r0);
    r0 = fmaf(W5[0], x5, r0); r0 = fmaf(W6[0], x6, r0);
    out0[o] = r0;

    float r1 = fmaf(W0[1], xv, Bc[1]);
    r1 = fmaf(W1[1], x1, r1); r1 = fmaf(W2[1], x2, r1);
    r1 = fmaf(W3[1], x3, r1); r1 = fmaf(W4[1], x4, r1);
    r1 = fmaf(W5[1], x5, r1); r1 = fmaf(W6[1], x6, r1);
    out1[o] = r1;
  }
}

extern "C" void kernel_launch(void* const* d_in, const int* in_sizes, int n_in,
                              void* d_out, int out_size, void* d_ws, size_t ws_size,
                              hipStream_t stream) {
  const float* X  = (const float*)d_in[0];
  const float* w0 = (const float*)d_in[1];
  const float* b0 = (const float*)d_in[2];
  const float* w1 = (const float*)d_in[3];
  const float* w2 = (const float*)d_in[4];
  const float* w3 = (const float*)d_in[5];
  const float* w4 = (const float*)d_in[6];
  const float* w5 = (const float*)d_in[7];
  const float* w6 = (const float*)d_in[8];
  float* out = (float*)d_out;

  dim3 grid(IMG / TW, IMG / TH, 32);
  fused_stencil<<<grid, dim3(NT), 0, stream>>>(X, w0, b0, w1, w2, w3, w4, w5, w6, out);
}